// SrcNGramRepeatBlock_71751723647497
// MI455X (gfx1250) — compile-verified
//
#include <hip/hip_runtime.h>

// SrcNGramRepeatBlock (n=3) — memory-bound streaming kernel.
//
// Roofline: 32MB token read + 8MB mask read + 32MB int32 write ≈ 72MB.
// At 23.3 TB/s HBM ≈ 3.1us floor; working set < 192MB L2, so replays are
// L2-resident. No matmul -> WMMA not applicable. The CDNA5-specific path used
// is the gfx1250 async global->LDS copy (GLOBAL_LOAD_ASYNC_TO_LDS_B128/B32,
// ASYNCcnt + s_wait_asynccnt): each 4KB token tile (+halo) and 1KB mask tile
// is staged in LDS once, so the 3-token sliding windows read LDS instead of
// re-fetching global memory.

constexpr int kBsz    = 1024;
constexpr int kSrcLen = 8192;
constexpr int kStep   = 2048;
constexpr int kN      = 3;
constexpr int kNw     = kSrcLen - kN + 1;   // 8190
constexpr int TILE    = 1024;               // tokens per block
constexpr int THREADS = 256;                // 8 wave32s

__device__ __forceinline__ unsigned lds_addr_of(const void* p) {
  // __shared__ pointer -> flat address; flat-LDS aperture keeps the LDS byte
  // offset in addr[31:0], so truncation yields the DS address for async ops.
  return (unsigned)(unsigned long long)p;
}

__global__ __launch_bounds__(THREADS)
void ngram_block_kernel(const int* __restrict__ orig,
                        const int* __restrict__ prev,
                        const unsigned char* __restrict__ mask,
                        const int* __restrict__ pad_ptr,
                        int* __restrict__ out) {
  __shared__ alignas(16) int           s_tok[TILE + 8];   // tile + halo tokens
  __shared__ alignas(16) unsigned char s_msk[TILE + 16];  // tile + halo mask

  const int tilesPerRow = kSrcLen / TILE;
  const int b         = blockIdx.x / tilesPerRow;
  const int tileStart = (blockIdx.x % tilesPerRow) * TILE;
  const int tid       = threadIdx.x;
  const bool hasHalo  = (tileStart + TILE) < kSrcLen;  // last tile needs none

  const int*           gtok = orig + (size_t)b * kSrcLen + tileStart;
  const unsigned char* gmsk = mask + (size_t)b * kSrcLen + tileStart;

  // ---- async global -> LDS staging (gfx1250 ASYNCcnt path) ----
  {
    unsigned ldst = lds_addr_of(s_tok) + (unsigned)(tid * 16);
    unsigned voff = (unsigned)(tid * 16);          // 256 lanes x 16B = 4KB
    asm volatile("global_load_async_to_lds_b128 %0, %1, %2"
                 :: "v"(ldst), "v"(voff), "s"(gtok) : "memory");
  }
  if (tid < (TILE / 16)) {                         // 64 lanes x 16B = 1KB mask
    unsigned ldst = lds_addr_of(s_msk) + (unsigned)(tid * 16);
    unsigned voff = (unsigned)(tid * 16);
    asm volatile("global_load_async_to_lds_b128 %0, %1, %2"
                 :: "v"(ldst), "v"(voff), "s"(gmsk) : "memory");
  }
  if (hasHalo && tid == 0) {                       // 2-token / 2-byte halo
    unsigned ldst = lds_addr_of(s_tok) + (unsigned)(TILE * 4);
    unsigned voff = (unsigned)(TILE * 4);
    asm volatile("global_load_async_to_lds_b128 %0, %1, %2"
                 :: "v"(ldst), "v"(voff), "s"(gtok) : "memory");
    unsigned mldst = lds_addr_of(s_msk) + (unsigned)TILE;
    unsigned mvoff = (unsigned)TILE;
    asm volatile("global_load_async_to_lds_b32 %0, %1, %2"
                 :: "v"(mldst), "v"(mvoff), "s"(gmsk) : "memory");
  }
  asm volatile("s_wait_asynccnt 0" ::: "memory");
  __syncthreads();

  // Suffix of prev_tokens (uniform per block) and pad scalar.
  const int s0  = prev[(size_t)b * kStep + (kStep - 2)];
  const int s1  = prev[(size_t)b * kStep + (kStep - 1)];
  const int pad = pad_ptr[0];

  const int base = tid * 4;
  int vals[4];
#pragma unroll
  for (int c = 0; c < 4; ++c) {
    const int j  = tileStart + base + c;
    const int t0 = s_tok[base + c];
    const int t1 = s_tok[base + c + 1];
    const int t2 = s_tok[base + c + 2];
    const bool match = (t0 == s0) && (t1 == s1);
    const bool prot  = s_msk[base + c + 2] != 0;
    const bool valid = j < kNw;                    // tail positions -> pad
    vals[c] = (valid && match && !prot) ? t2 : pad;
  }
  int4 r;
  r.x = vals[0]; r.y = vals[1]; r.z = vals[2]; r.w = vals[3];
  *reinterpret_cast<int4*>(out + (size_t)b * kSrcLen + tileStart + base) = r;
}

extern "C" void kernel_launch(void* const* d_in, const int* in_sizes, int n_in,
                              void* d_out, int out_size, void* d_ws, size_t ws_size,
                              hipStream_t stream) {
  const int*           orig = (const int*)d_in[0];
  const int*           prev = (const int*)d_in[1];
  const unsigned char* mask = (const unsigned char*)d_in[2];
  // d_in[3] = n (==3, structural), d_in[4] = vocab_size (unused by reference),
  // d_in[5] = pad (read on-device to stay faithful).
  const int* pad_ptr = (const int*)d_in[5];
  int* out = (int*)d_out;
  (void)in_sizes; (void)n_in; (void)out_size; (void)d_ws; (void)ws_size;

  const int blocks = kBsz * (kSrcLen / TILE);  // 8192 blocks x 256 threads
  ngram_block_kernel<<<blocks, THREADS, 0, stream>>>(orig, prev, mask, pad_ptr, out);
}